// get_loss_50002009260620
// MI455X (gfx1250) — compile-verified
//
#include <hip/hip_runtime.h>
#include <hip/hip_bf16.h>

typedef __attribute__((ext_vector_type(2))) float v2f;
typedef __attribute__((ext_vector_type(8))) float v8f;

#define NUM_B 32
#define NUM_P 4096
#define NUM_N 16384
#define NUM_C 14

// ---------------------------------------------------------------------------
// Kernel 1: per-batch GT centroids (class means) + chamfer + separation loss.
// One block per batch. LDS atomics over P=4096 points, then 14x14 pair matrix.
// ---------------------------------------------------------------------------
__global__ void centroid_chamfer_kernel(const float* __restrict__ origin,   // [B,9,P]
                                        const int*   __restrict__ target,   // [B,P]
                                        const float* __restrict__ cpred,    // [B,3,C]
                                        float* __restrict__ cent,           // ws [B,C,3]
                                        float* __restrict__ c2,             // ws [B,C]
                                        float* __restrict__ out)
{
    __shared__ float s_sum[NUM_C * 3];
    __shared__ float s_cnt[NUM_C];
    __shared__ float s_cent[NUM_C * 3];
    __shared__ float s_D[NUM_C * NUM_C];

    const int b   = blockIdx.x;
    const int tid = threadIdx.x;

    if (tid < NUM_C * 3) s_sum[tid] = 0.0f;
    if (tid < NUM_C)     s_cnt[tid] = 0.0f;
    __syncthreads();

    // Accumulate class sums of xyz (rows 0..2 of origin_points).
    const float* ob = origin + (size_t)b * 9 * NUM_P;
    for (int p = tid; p < NUM_P; p += blockDim.x) {
        int t = target[b * NUM_P + p];
        t = min(max(t, 0), NUM_C - 1);
        atomicAdd(&s_sum[t * 3 + 0], ob[p]);
        atomicAdd(&s_sum[t * 3 + 1], ob[NUM_P + p]);
        atomicAdd(&s_sum[t * 3 + 2], ob[2 * NUM_P + p]);
        atomicAdd(&s_cnt[t], 1.0f);
    }
    __syncthreads();

    if (tid < NUM_C * 3) {
        int c = tid / 3;
        float v = s_sum[tid] / fmaxf(s_cnt[c], 1.0f);
        s_cent[tid] = v;
        cent[b * NUM_C * 3 + tid] = v;
    }
    __syncthreads();

    if (tid < NUM_C) {
        float x = s_cent[tid * 3 + 0];
        float y = s_cent[tid * 3 + 1];
        float z = s_cent[tid * 3 + 2];
        c2[b * NUM_C + tid] = x * x + y * y + z * z;
    }

    // Chamfer pair matrix D[pred i][gt j].
    if (tid < NUM_C * NUM_C) {
        int i = tid / NUM_C, j = tid % NUM_C;
        float dx = cpred[b * 3 * NUM_C + 0 * NUM_C + i] - s_cent[j * 3 + 0];
        float dy = cpred[b * 3 * NUM_C + 1 * NUM_C + i] - s_cent[j * 3 + 1];
        float dz = cpred[b * 3 * NUM_C + 2 * NUM_C + i] - s_cent[j * 3 + 2];
        s_D[tid] = dx * dx + dy * dy + dz * dz;
    }
    __syncthreads();

    if (tid < NUM_C) {
        // l1 term: min over pred rows for column tid.
        float cm = 1e30f;
        for (int i = 0; i < NUM_C; ++i) cm = fminf(cm, s_D[i * NUM_C + tid]);
        // l2 term + separation (two smallest euclidean dists) for row tid.
        float rm = 1e30f;
        float b0 = 1e30f, b1 = 1e30f;
        for (int j = 0; j < NUM_C; ++j) {
            float d = s_D[tid * NUM_C + j];
            rm = fminf(rm, d);
            float dd = sqrtf(d);
            if (dd < b0) { b1 = b0; b0 = dd; }
            else if (dd < b1) { b1 = dd; }
        }
        atomicAdd(out, cm + rm + 0.1f * (b0 / b1));
    }
}

// ---------------------------------------------------------------------------
// Kernel 2: distance-estimate loss via V_WMMA_F32_16X16X4_F32.
// A = centroids (M=class, 14 real + 2 zero rows), B = 16-point tile (N=point).
// D: VGPR r = class (r + 8*upperhalf), lane = point. So each lane owns 8 of
// its point's class dot-products in registers: the min over classes is an
// in-register chain plus ONE xor-16 shuffle to merge half-wave class ranges.
// d^2 = |s|^2 + min_c(|c|^2 - 2 s.c); smooth-L1 against displacement_pred.
// ---------------------------------------------------------------------------
__global__ void distance_loss_kernel(const float* __restrict__ disp,   // [B,N]
                                     const float* __restrict__ sub,    // [B,3,N]
                                     const float* __restrict__ cent,   // [B,C,3]
                                     const float* __restrict__ c2,     // [B,C]
                                     float* __restrict__ out)
{
    const int SPLIT = 8;                      // blocks per batch
    const int b     = blockIdx.x / SPLIT;
    const int split = blockIdx.x % SPLIT;
    const int lane  = threadIdx.x & 31;
    const int wave  = threadIdx.x >> 5;
    const int wavesPerBlock = blockDim.x >> 5;            // 8
    const int waveSlot      = split * wavesPerBlock + wave;
    const int wavesPerBatch = SPLIT * wavesPerBlock;      // 64

    const int  n  = lane & 15;        // class row for A, point index for B
    const bool up = lane >= 16;

    // A matrix (centroids), constant per wave. Rows 14,15 are zero.
    v2f am;
    if (n < NUM_C) {
        const float* cb = cent + (b * NUM_C + n) * 3;
        am.x = up ? cb[2] : cb[0];      // K2 : K0
        am.y = up ? 0.0f : cb[1];       // K3 : K1
    } else {
        am.x = 0.0f; am.y = 0.0f;
    }

    // Per-register class |c|^2: lower half-wave sees classes 0-7 in VGPRs
    // 0-7, upper half-wave sees classes 8-15 (14,15 padded to +inf-ish).
    float c2r[8];
#pragma unroll
    for (int r = 0; r < 8; ++r) {
        int c = r + (up ? 8 : 0);
        c2r[r] = (c < NUM_C) ? c2[b * NUM_C + c] : 1e30f;
    }

    const float* sx = sub  + (size_t)b * 3 * NUM_N;
    const float* dv = disp + (size_t)b * NUM_N;
    const int tiles = NUM_N / 16;      // 1024, uniform 16 iters per wave

    float acc = 0.0f;
    for (int tile = waveSlot; tile < tiles; tile += wavesPerBatch) {
        const int p = tile * 16 + n;
        float px = sx[p];
        float py = sx[NUM_N + p];
        float pz = sx[2 * NUM_N + p];
        float dp = dv[p];
        float s2 = px * px + py * py + pz * pz;

        v2f bm;
        bm.x = up ? pz : px;            // K2 : K0
        bm.y = up ? 0.0f : py;          // K3 : K1

        v8f d = {};
        d = __builtin_amdgcn_wmma_f32_16x16x4_f32(
                /*neg_a=*/false, am, /*neg_b=*/false, bm,
                /*c_mod=*/(short)0, d, /*reuse_a=*/false, /*reuse_b=*/false);

        // In-register min over this half-wave's 8 classes.
        float m = fmaf(-2.0f, d[0], c2r[0]);
#pragma unroll
        for (int r = 1; r < 8; ++r)
            m = fminf(m, fmaf(-2.0f, d[r], c2r[r]));
        // Merge classes 0-7 with 8-15: single cross-half shuffle.
        m = fminf(m, __shfl_xor(m, 16, 32));

        // Raw v_sqrt_f32 (<=1 ulp); operands are O(1), no denormal concerns.
        float dist = __builtin_amdgcn_sqrtf(fmaxf(s2 + m, 0.0f));
        float x  = dp - dist;
        float ax = fabsf(x);
        // Both half-waves hold bit-identical losses; accumulate on all lanes
        // and halve at the end (exact: pair-sum a+a=2a, *0.5 is exact).
        acc += (ax < 1.0f) ? 0.5f * x * x : (ax - 0.5f);
    }

    // Reduce all 32 lanes (cross-half pairs first), one atomic per wave.
    acc += __shfl_xor(acc, 16, 32);
    acc += __shfl_xor(acc, 1, 32);
    acc += __shfl_xor(acc, 2, 32);
    acc += __shfl_xor(acc, 4, 32);
    acc += __shfl_xor(acc, 8, 32);
    if (lane == 0) atomicAdd(out, 0.5f * acc);
}

// ---------------------------------------------------------------------------
extern "C" void kernel_launch(void* const* d_in, const int* in_sizes, int n_in,
                              void* d_out, int out_size, void* d_ws, size_t ws_size,
                              hipStream_t stream)
{
    (void)in_sizes; (void)n_in; (void)ws_size;
    const float* disp   = (const float*)d_in[0];   // [B,N]
    const float* sub    = (const float*)d_in[1];   // [B,3,N]
    const float* cpred  = (const float*)d_in[2];   // [B,3,C]
    const float* origin = (const float*)d_in[3];   // [B,9,P]
    const int*   target = (const int*)d_in[4];     // [B,P]
    float* out = (float*)d_out;

    float* cent = (float*)d_ws;                    // B*C*3 floats
    float* c2   = cent + NUM_B * NUM_C * 3;        // B*C floats

    hipMemsetAsync(d_out, 0, sizeof(float) * (size_t)out_size, stream);

    centroid_chamfer_kernel<<<NUM_B, 256, 0, stream>>>(origin, target, cpred,
                                                       cent, c2, out);
    distance_loss_kernel<<<NUM_B * 8, 256, 0, stream>>>(disp, sub, cent, c2, out);
}